// MriSenseNufft_39213051412566
// MI455X (gfx1250) — compile-verified
//
#include <hip/hip_runtime.h>
#include <hip/hip_bf16.h>

// MriSenseNufft forward on gfx1250 (MI455X).
// DFT-as-complex-GEMM on the fp32 WMMA pipe (V_WMMA_F32_16X16X4_F32),
// software-pipelined fragment loads, KB interpolation as an L2-resident
// gather kernel.

typedef float v2f __attribute__((ext_vector_type(2)));
typedef float v8f __attribute__((ext_vector_type(8)));

// Explicit global address space: guarantees global_load/global_store lowering.
typedef const float __attribute__((address_space(1)))* gcf;
typedef float __attribute__((address_space(1)))* gf;

#define PI_F 3.14159265358979323846f
#define BB 2
#define CC 8
#define MM 65536
#define NIM 256          // image size per dim
#define KG 512           // oversampled grid per dim
#define NPIX (NIM * NIM) // 65536
#define GPIX (KG * KG)   // 262144
#define JW 6
#define TLEN 6144        // J*L
#define WAVES_PER_BLOCK 4
#define MT 4             // row-tiles (16 rows each) per wave -> 64x16 macro-tile

// ---------------------------------------------------------------------------
// Build DFT twiddle matrices.
//   Wc : 512x256, Wc[k][n] = exp(-2*pi*i*k*n/512) / 512   (1/512 = fftn norm)
//   Wt : 256x512, Wt[n][k] = exp(-2*pi*i*n*k/512)
// ---------------------------------------------------------------------------
__global__ void build_twiddles(float* __restrict__ wcre, float* __restrict__ wcim,
                               float* __restrict__ wtre, float* __restrict__ wtim) {
  int t = blockIdx.x * blockDim.x + threadIdx.x;
  if (t >= KG * NIM) return;
  float s, c;
  {
    int k = t >> 8, n = t & 255;                       // Wc: [k(512)][n(256)]
    float ang = -2.0f * PI_F * (float)((k * n) & 511) * (1.0f / 512.0f);
    __sincosf(ang, &s, &c);
    wcre[t] = c * (1.0f / 512.0f);
    wcim[t] = s * (1.0f / 512.0f);
  }
  {
    int n = t >> 9, k = t & 511;                       // Wt: [n(256)][k(512)]
    float ang = -2.0f * PI_F * (float)((n * k) & 511) * (1.0f / 512.0f);
    __sincosf(ang, &s, &c);
    wtre[t] = c;
    wtim[t] = s;
  }
}

// ---------------------------------------------------------------------------
// Apodization: img[b,c] = (x[b] * smap[b,c]) * sn   (complex multiply, fp32)
// ---------------------------------------------------------------------------
__global__ void apodize(const float* __restrict__ x, const float* __restrict__ smap,
                        const float* __restrict__ sn,
                        float* __restrict__ ire, float* __restrict__ iim) {
  int t = blockIdx.x * blockDim.x + threadIdx.x;
  if (t >= BB * CC * NPIX) return;
  int p  = t & (NPIX - 1);
  int bc = t >> 16;        // 0..15
  int b  = bc >> 3;        // /C
  float xr = x[(size_t)b * 2 * NPIX + p];
  float xi = x[(size_t)b * 2 * NPIX + NPIX + p];
  const float* s = smap + (size_t)bc * 2 * NPIX;
  float sr = s[p];
  float si = s[p + NPIX];
  float a  = sn[p];
  ire[t] = (xr * sr - xi * si) * a;
  iim[t] = (xr * si + xi * sr) * a;
}

// ---------------------------------------------------------------------------
// Batched complex GEMM on the fp32 WMMA pipe:  C = A x B  (row-major)
//   A: Mdim x Kdim complex, B: Kdim x Ndim complex, C: Mdim x Ndim complex.
// Each wave computes a 64x16 macro-tile (MT=4 row-tiles x 1 col-tile): the
// strided B fragment (4x b32 loads / k-step) is reused by 4 row-tiles, giving
// 16 WMMAs per k-step against 12 loads.  Fragment loads are software-
// pipelined one k-step ahead so the s_wait for step k+1's data sits behind
// the 16 independent WMMAs of step k.
// F32 WMMA has no A/B NEG modifier (ISA NEG table); the complex cross term
// -Aim*Bim is realized by negating the SHARED Bim fragment (2 xors / step).
// Fragment layouts per CDNA5 ISA (16x16x4 f32):
//   A: lane l(0..15) holds A[row+l][k+2*half .. +1]   (half = lane>>4)
//   B: v0/v1 hold rows k+2*half, k+2*half+1 at col0+l16
//   C: vgpr r -> row row0 + r + 8*half, col col0 + l16
// ---------------------------------------------------------------------------
__global__ void cgemm_dft_wmma(const float* __restrict__ Are, const float* __restrict__ Aim,
                               const float* __restrict__ Bre, const float* __restrict__ Bim,
                               float* __restrict__ Cre, float* __restrict__ Cim,
                               int Mdim, int Ndim, int Kdim,
                               long aB, long bB, long cB) {
  const int wave = threadIdx.x >> 5;
  const int lane = threadIdx.x & 31;
  const int half = lane >> 4;
  const int l16  = lane & 15;

  const int tilesN = Ndim >> 4;
  const int tilesTotal = (Mdim >> 6) * tilesN;     // 64-row macro tiles
  const int tile = blockIdx.x * WAVES_PER_BLOCK + wave;
  if (tile >= tilesTotal) return;                  // whole-wave uniform: EXEC stays all-1

  const int row0 = (tile / tilesN) << 6;
  const int col0 = (tile % tilesN) << 4;
  const long batch = blockIdx.y;

  gcf are = (gcf)(Are + batch * aB);
  gcf aim = (gcf)(Aim + batch * aB);
  gcf bre = (gcf)(Bre + batch * bB);
  gcf bim = (gcf)(Bim + batch * bB);

  // Loop-carried pointers: A advances 4 floats / step, B advances 4 rows / step.
  gcf par[MT];
  gcf pai[MT];
#pragma unroll
  for (int r = 0; r < MT; ++r) {
    const size_t ro = (size_t)(row0 + 16 * r + l16) * (size_t)Kdim + 2 * half;
    par[r] = are + ro;
    pai[r] = aim + ro;
  }
  const size_t bo = (size_t)(2 * half) * Ndim + col0 + l16;
  gcf pbr = bre + bo;
  gcf pbi = bim + bo;
  const size_t bstep = 4 * (size_t)Ndim;

  v8f accre[MT];
  v8f accim[MT];
#pragma unroll
  for (int r = 0; r < MT; ++r) { accre[r] = {}; accim[r] = {}; }

  // --- software pipeline: prologue loads for k = 0 ---
  v2f br = { pbr[0], pbr[Ndim] };
  v2f bi = { pbi[0], pbi[Ndim] };
  v2f ar[MT], ai[MT];
#pragma unroll
  for (int r = 0; r < MT; ++r) {
    ar[r] = (v2f){ par[r][0], par[r][1] };
    ai[r] = (v2f){ pai[r][0], pai[r][1] };
  }

  for (int k = 0; k < Kdim - 4; k += 4) {
    // issue next step's loads before this step's WMMAs
    pbr += bstep;
    pbi += bstep;
    v2f brn = { pbr[0], pbr[Ndim] };
    v2f bin = { pbi[0], pbi[Ndim] };
    v2f arn[MT], ain[MT];
#pragma unroll
    for (int r = 0; r < MT; ++r) {
      par[r] += 4;
      pai[r] += 4;
      arn[r] = (v2f){ par[r][0], par[r][1] };
      ain[r] = (v2f){ pai[r][0], pai[r][1] };
    }

    v2f bneg = { -bi[0], -bi[1] };         // shared negation, 2 xors / k-step
#pragma unroll
    for (int r = 0; r < MT; ++r) {
      // re += ar*br + ai*(-bi) ;  im += ar*bi + ai*br
      accre[r] = __builtin_amdgcn_wmma_f32_16x16x4_f32(false, ar[r], false, br,   (short)0, accre[r], false, false);
      accre[r] = __builtin_amdgcn_wmma_f32_16x16x4_f32(false, ai[r], false, bneg, (short)0, accre[r], false, false);
      accim[r] = __builtin_amdgcn_wmma_f32_16x16x4_f32(false, ar[r], false, bi,   (short)0, accim[r], false, false);
      accim[r] = __builtin_amdgcn_wmma_f32_16x16x4_f32(false, ai[r], false, br,   (short)0, accim[r], false, false);
    }

    br = brn;
    bi = bin;
#pragma unroll
    for (int r = 0; r < MT; ++r) { ar[r] = arn[r]; ai[r] = ain[r]; }
  }

  // --- epilogue: last k-step ---
  {
    v2f bneg = { -bi[0], -bi[1] };
#pragma unroll
    for (int r = 0; r < MT; ++r) {
      accre[r] = __builtin_amdgcn_wmma_f32_16x16x4_f32(false, ar[r], false, br,   (short)0, accre[r], false, false);
      accre[r] = __builtin_amdgcn_wmma_f32_16x16x4_f32(false, ai[r], false, bneg, (short)0, accre[r], false, false);
      accim[r] = __builtin_amdgcn_wmma_f32_16x16x4_f32(false, ar[r], false, bi,   (short)0, accim[r], false, false);
      accim[r] = __builtin_amdgcn_wmma_f32_16x16x4_f32(false, ai[r], false, br,   (short)0, accim[r], false, false);
    }
  }

  gf cre = (gf)(Cre + batch * cB);
  gf cim = (gf)(Cim + batch * cB);
#pragma unroll
  for (int r = 0; r < MT; ++r) {
#pragma unroll
    for (int i = 0; i < 8; ++i) {
      const int row = row0 + 16 * r + i + 8 * half;
      cre[(size_t)row * Ndim + col0 + l16] = accre[r][i];
      cim[(size_t)row * Ndim + col0 + l16] = accim[r][i];
    }
  }
}

// ---------------------------------------------------------------------------
// Kaiser-Bessel table interpolation + phase ramp.
// One thread per (b, m): computes separable 6+6 weights/indices once,
// then gathers 6x6 neighborhoods for all 8 coils (grid is L2-resident).
// out: (B, C, 2, M)
// ---------------------------------------------------------------------------
__global__ void interp_kb(const float* __restrict__ Gre, const float* __restrict__ Gim,
                          const float* __restrict__ om,
                          const float* __restrict__ tab0, const float* __restrict__ tab1,
                          float* __restrict__ out) {
  int t = blockIdx.x * blockDim.x + threadIdx.x;
  if (t >= BB * MM) return;
  const int b = t >> 16;
  const int m = t & (MM - 1);

  const float o0 = om[(size_t)(b * 2 + 0) * MM + m];
  const float o1 = om[(size_t)(b * 2 + 1) * MM + m];

  gcf t0 = (gcf)tab0;
  gcf t1 = (gcf)tab1;

  float w0[JW], w1[JW];
  int g0[JW], g1[JW];
  {
    const float tm = o0 * ((float)KG / (2.0f * PI_F));
    const float koff = floorf(tm - 0.5f * JW);
#pragma unroll
    for (int i = 0; i < JW; ++i) {
      const float kk = koff + (float)(i + 1);
      const float dist = tm - kk;
      int tidx = (int)roundf((dist + 0.5f * JW) * 1024.0f);
      tidx = min(max(tidx, 0), TLEN);
      w0[i] = t0[tidx];
      const int ik = (int)kk;
      g0[i] = (((ik + 1024) & (KG - 1)) << 9);   // pre-multiplied row offset
    }
  }
  {
    const float tm = o1 * ((float)KG / (2.0f * PI_F));
    const float koff = floorf(tm - 0.5f * JW);
#pragma unroll
    for (int i = 0; i < JW; ++i) {
      const float kk = koff + (float)(i + 1);
      const float dist = tm - kk;
      int tidx = (int)roundf((dist + 0.5f * JW) * 1024.0f);
      tidx = min(max(tidx, 0), TLEN);
      w1[i] = t1[tidx];
      const int ik = (int)kk;
      g1[i] = (ik + 1024) & (KG - 1);
    }
  }

  float ps, pc;
  __sincosf((o0 + o1) * 128.0f, &ps, &pc);   // N_SHIFT = (128, 128)

  const size_t base = (size_t)b * CC * GPIX;
  for (int c = 0; c < CC; ++c) {
    gcf gr = (gcf)(Gre + base + (size_t)c * GPIX);
    gcf gi = (gcf)(Gim + base + (size_t)c * GPIX);
    float yr = 0.0f, yi = 0.0f;
#pragma unroll
    for (int i = 0; i < JW; ++i) {
      const float wi = w0[i];
      const int ro = g0[i];
#pragma unroll
      for (int j = 0; j < JW; ++j) {
        const float w = wi * w1[j];
        const int off = ro + g1[j];
        yr += gr[off] * w;
        yi += gi[off] * w;
      }
    }
    const float zr = yr * pc - yi * ps;
    const float zi = yr * ps + yi * pc;
    const size_t ob = (size_t)(b * CC + c) * 2 * MM + m;
    gf po = (gf)(out + ob);
    po[0]  = zr;
    po[MM] = zi;
  }
}

// ---------------------------------------------------------------------------
// Workspace layout (floats):
//   Wc re/im : 2 x 131072      Wt re/im : 2 x 131072
//   img re/im: 2 x 1048576     T re/im  : 2 x 2097152     G re/im: 2 x 4194304
// Total 15,204,352 floats = ~58 MB.
// ---------------------------------------------------------------------------
extern "C" void kernel_launch(void* const* d_in, const int* in_sizes, int n_in,
                              void* d_out, int out_size, void* d_ws, size_t ws_size,
                              hipStream_t stream) {
  const float* x    = (const float*)d_in[0];
  const float* smap = (const float*)d_in[1];
  const float* om   = (const float*)d_in[2];
  const float* sn   = (const float*)d_in[3];
  const float* tab0 = (const float*)d_in[4];
  const float* tab1 = (const float*)d_in[5];
  float* out = (float*)d_out;

  float* ws   = (float*)d_ws;
  float* wcre = ws;
  float* wcim = ws + 131072;
  float* wtre = ws + 262144;
  float* wtim = ws + 393216;
  float* ire  = ws + 524288;
  float* iim  = ws + 1572864;
  float* tre  = ws + 2621440;
  float* tim  = ws + 4718592;
  float* gre  = ws + 6815744;
  float* gim  = ws + 11010048;

  build_twiddles<<<512, 256, 0, stream>>>(wcre, wcim, wtre, wtim);
  apodize<<<4096, 256, 0, stream>>>(x, smap, sn, ire, iim);

  // Stage 1: T(512x256) = Wc(512x256) x img(256x256), 16 batches share Wc.
  {
    const int tiles = (KG / 64) * (NIM / 16);       // 128 macro-tiles
    dim3 grid(tiles / WAVES_PER_BLOCK, BB * CC);
    cgemm_dft_wmma<<<grid, 32 * WAVES_PER_BLOCK, 0, stream>>>(
        wcre, wcim, ire, iim, tre, tim,
        KG, NIM, NIM,
        0L, (long)NPIX, (long)(KG * NIM));
  }
  // Stage 2: G(512x512) = T(512x256) x Wt(256x512), 16 batches share Wt.
  {
    const int tiles = (KG / 64) * (KG / 16);        // 256 macro-tiles
    dim3 grid(tiles / WAVES_PER_BLOCK, BB * CC);
    cgemm_dft_wmma<<<grid, 32 * WAVES_PER_BLOCK, 0, stream>>>(
        tre, tim, wtre, wtim, gre, gim,
        KG, KG, NIM,
        (long)(KG * NIM), 0L, (long)GPIX);
  }

  interp_kb<<<512, 256, 0, stream>>>(gre, gim, om, tab0, tab1, out);
}